// RoIPooling_87943750352913
// MI455X (gfx1250) — compile-verified
//
#include <hip/hip_runtime.h>
#include <cstdint>

// Problem constants (fixed by setup_inputs: B=4, C=512, H=W=50, R=128).
#define BATCH  4
#define CHANS  512
#define HH     50
#define WW     50
#define RR     128
#define NROI   (BATCH * RR)       // 512 rois
#define HW     (HH * WW)          // 2500
#define CC     4                  // channels staged per LDS chunk
#define CSPLIT 8                  // channel slices across gridDim.y
#define CPB    (CHANS / CSPLIT)   // 64 channels per block
#define NCHUNK (CPB / CC)         // 16 chunks
#define BUFSZ  (CC * HW)          // floats per buffer (worst case 10000 = 40 KB)
#define NEGV   (-1e30f)

// Force a wave-uniform value into the SGPR register class for "s" constraints
// (clang keeps v_med3_i32-derived uniforms in VGPRs otherwise).
#define RFL(x) ((unsigned)__builtin_amdgcn_readfirstlane((int)(x)))

typedef unsigned v4u __attribute__((ext_vector_type(4)));

__global__ __launch_bounds__(256)
void roipool_tdm_kernel(const float* __restrict__ img,
                        const int*   __restrict__ ids,
                        const float* __restrict__ rois,
                        float*       __restrict__ out)
{
    // Double-buffered staging: 2 x 4 x 50x50 fp32 = 80 KB (WGP has 320 KB).
    __shared__ __align__(16) float buf[2 * BUFSZ];

    const int n     = blockIdx.x;          // roi index (uniform)
    const int cbase = blockIdx.y * CPB;    // channel slice base (uniform)
    const int b     = ids[n >> 7];         // R = 128; uniform scalar load

    // ---- roi bounds, bit-exact vs the JAX reference (all scalar) ----
    const float4 r = reinterpret_cast<const float4*>(rois)[n];
    const float scale = 1.0f / 16.0f;      // 1/REDUCTION
    const int x1 = (int)rintf(r.x * scale);   // jnp.round == round-half-even
    const int y1 = (int)rintf(r.y * scale);
    const int x2 = (int)rintf(r.z * scale);
    const int y2 = (int)rintf(r.w * scale);
    const float roi_w = (float)max(x2 - x1 + 1, 1);
    const float roi_h = (float)max(y2 - y1 + 1, 1);
    const float bin_h = roi_h / 7.0f;
    const float bin_w = roi_w / 7.0f;
    auto clipH = [](int v) { return min(max(v, 0), HH); };
    auto clipW = [](int v) { return min(max(v, 0), WW); };

    // Rows/cols the roi can ever touch: [r0,r1) x [w0,w1)
    const int r0 = clipH(y1);
    const int r1 = clipH((int)ceilf(7.0f * bin_h) + y1);
    const int w0 = clipW(x1);
    const int w1 = clipW((int)ceilf(7.0f * bin_w) + x1);
    const int nrows = r1 - r0;
    const int w0e   = w0 & ~1;             // even-aligned column window
    const int w1e   = (w1 + 1) & ~1;       // <= 50, stays inside the row
    const int pitch = w1e - w0e;           // even, <= 50

    const uint64_t gbase   = (uint64_t)(uintptr_t)(img + (size_t)b * (size_t)CHANS * HW);
    const unsigned ldsBase = (unsigned)(uintptr_t)(void*)buf;  // low 32b of flat = LDS offset

    const int tid = threadIdx.x;

    // ---- per-thread output bin (one bin per thread per chunk) ----
    const bool active = tid < CC * 49;
    int c_l = 0, hs = 0, he = 0, ws = 0, we = 0;
    if (active) {
        c_l = tid / 49;
        const int bin = tid % 49;
        const int ph = bin / 7, pw = bin % 7;
        hs = clipH((int)floorf((float)ph       * bin_h) + y1);
        he = clipH((int)ceilf ((float)(ph + 1) * bin_h) + y1);
        ws = clipW((int)floorf((float)pw       * bin_w) + x1);
        we = clipW((int)ceilf ((float)(pw + 1) * bin_w) + x1);
    }

    // ---- Tensor DMA descriptor group 1 words (chunk-invariant, uniform) ----
    // 3D tile: X = pitch contiguous floats, Y = nrows (stride 50),
    //          Z = CC channels (stride 2500). LDS dest is fully packed.
    const unsigned q10 = RFL(0x00020000u);                          // wg_mask=0, data_size=2 (4B)
    const unsigned q11 = RFL((unsigned)pitch << 16);                // tensor_dim0[15:0]
    const unsigned q12 = RFL((unsigned)nrows << 16);                // dim0 hi=0 | tensor_dim1[15:0]
    const unsigned q13 = RFL((unsigned)pitch << 16);                // dim1 hi=0 | tile_dim0
    const unsigned q14 = RFL((unsigned)nrows | ((unsigned)CC << 16)); // tile_dim1 | tile_dim2
    const unsigned q15 = RFL((unsigned)WW);                         // tensor_dim0_stride lo
    const unsigned q16 = RFL((unsigned)HW << 16);                   // stride0 hi | dim1_stride[15:0]
    const unsigned q17 = RFL(0u);                                   // dim1_stride[47:16]
    v4u g2; g2[0] = RFL((unsigned)CC); g2[1] = 0u; g2[2] = 0u; g2[3] = 0u;  // tensor_dim2
    v4u g3; g3[0] = 0u; g3[1] = 0u; g3[2] = 0u; g3[3] = 0u;

    // Wave 0 issues one TDM op per chunk (EXEC-independent, per-wave).
    // Group 1 (256-bit) is placed in a fixed quad-aligned octet s[40:47] via
    // s_mov, since clang's "s" constraint only allocates SGPRs up to 128 bits.
    auto stage = [&](int chunk) {
        const int c0 = cbase + chunk * CC;
        const uint64_t ga = gbase + (uint64_t)(unsigned)((c0 * HW + r0 * WW + w0e) * 4);
        v4u g0;
        g0[0] = RFL(1u);                                            // count=1, is_restore=0
        g0[1] = RFL(ldsBase + (unsigned)((chunk & 1) * BUFSZ * 4)); // lds_addr
        g0[2] = RFL((unsigned)ga);                                  // global_addr[31:0]
        g0[3] = RFL(((unsigned)(ga >> 32) & 0x01FFFFFFu)            // global_addr[56:32]
                    | 0x80000000u);                                 // type=2 ("image")
        asm volatile(
            "s_mov_b32 s40, %3\n\t"
            "s_mov_b32 s41, %4\n\t"
            "s_mov_b32 s42, %5\n\t"
            "s_mov_b32 s43, %6\n\t"
            "s_mov_b32 s44, %7\n\t"
            "s_mov_b32 s45, %8\n\t"
            "s_mov_b32 s46, %9\n\t"
            "s_mov_b32 s47, %10\n\t"
            "tensor_load_to_lds %0, s[40:47], %1, %2"
            :
            : "s"(g0), "s"(g2), "s"(g3),
              "s"(q10), "s"(q11), "s"(q12), "s"(q13),
              "s"(q14), "s"(q15), "s"(q16), "s"(q17)
            : "s40", "s41", "s42", "s43", "s44", "s45", "s46", "s47",
              "memory");
    };

    if (tid < 32) stage(0);  // prologue: chunk 0 in flight

    for (int chunk = 0; chunk < NCHUNK; ++chunk) {
        // Wave 0 has the only nonzero TENSORcnt; wait is a no-op elsewhere.
        asm volatile("s_wait_tensorcnt 0x0" ::: "memory");
        __syncthreads();

        // Kick off the next chunk's DMA; it overlaps with compute below. The
        // buffer it overwrites was last read in compute(chunk-1), which all
        // waves finished before the barrier above.
        if (chunk + 1 < NCHUNK && tid < 32) stage(chunk + 1);

        // ---- max-pool this chunk out of LDS; coalesced store ----
        if (active) {
            const float* bp = &buf[(chunk & 1) * BUFSZ];
            float m = NEGV;
            for (int h = hs; h < he; ++h) {
                const int rb = (c_l * nrows + (h - r0)) * pitch - w0e;
                for (int w = ws; w < we; ++w)
                    m = fmaxf(m, bp[rb + w]);
            }
            const bool empty = (he <= hs) || (we <= ws);
            const int c0 = cbase + chunk * CC;
            // ((n*C + c0 + c_l)*49 + bin) == (n*C + c0)*49 + tid -> contiguous
            out[(size_t)(n * CHANS + c0) * 49 + (size_t)tid] = empty ? 0.0f : m;
        }
    }
}

extern "C" void kernel_launch(void* const* d_in, const int* in_sizes, int n_in,
                              void* d_out, int out_size, void* d_ws, size_t ws_size,
                              hipStream_t stream) {
    (void)in_sizes; (void)n_in; (void)d_ws; (void)ws_size; (void)out_size;
    const float* img  = (const float*)d_in[0];
    const int*   ids  = (const int*)d_in[1];
    const float* rois = (const float*)d_in[2];
    float*       out  = (float*)d_out;

    dim3 grid(NROI, CSPLIT);   // 512 x 8 = 4096 blocks, 8 wave32s each
    roipool_tdm_kernel<<<grid, dim3(256), 0, stream>>>(img, ids, rois, out);
}